// GConvElman_56985626083403
// MI455X (gfx1250) — compile-verified
//
#include <hip/hip_runtime.h>

typedef __attribute__((ext_vector_type(2))) float v2f;
typedef __attribute__((ext_vector_type(4))) float v4f;
typedef __attribute__((ext_vector_type(8))) float v8f;

#define D 64

// ---------------- zero workspace accumulators (128-bit stores) ----------------
__global__ void zero_kernel(v4f* __restrict__ p, long long n4) {
  long long i = (long long)blockIdx.x * blockDim.x + threadIdx.x;
  long long stride = (long long)gridDim.x * blockDim.x;
  v4f z = {0.f, 0.f, 0.f, 0.f};
  for (; i < n4; i += stride) p[i] = z;
}

// ---------------- edge scatter: agg[dst] += x[src] * ew ----------------
// One thread per (edge, 4-channel chunk): b128 gather + 4x global_atomic_add_f32.
__global__ __launch_bounds__(256)
void scatter_kernel(const float* __restrict__ x,
                    const int* __restrict__ src,
                    const int* __restrict__ dst,
                    const float* __restrict__ ew,
                    float* __restrict__ agg, int E) {
  const long long total = (long long)E * (D / 4);
  long long idx = (long long)blockIdx.x * blockDim.x + threadIdx.x;
  const long long stride = (long long)gridDim.x * blockDim.x;
  for (; idx < total; idx += stride) {
    const int e  = (int)(idx >> 4);        // 16 chunks of 4 channels per edge
    const int c4 = (int)(idx & 15) << 2;
    const float w = ew[e];
    const v4f xv = *(const v4f*)(x + (long long)src[e] * D + c4);
    float* a = agg + (long long)dst[e] * D + c4;
    unsafeAtomicAdd(a + 0, xv.x * w);
    unsafeAtomicAdd(a + 1, xv.y * w);
    unsafeAtomicAdd(a + 2, xv.z * w);
    unsafeAtomicAdd(a + 3, xv.w * w);
  }
}

// ---------------- fused: out = sigmoid(agg @ Wrel^T + xin @ Wroot^T + biases) ----------------
// One wave per band of 16 nodes; 4 output N-tiles; K=64 via 16x V_WMMA_F32_16X16X4_F32.
__global__ __launch_bounds__(256)
void gconv_wmma_kernel(const float* __restrict__ agg,    // [N,64]
                       const float* __restrict__ xin,    // [N,64]
                       const float* __restrict__ Wrel,   // [64,64] row-major
                       const float* __restrict__ Wroot,  // [64,64] row-major
                       const float* __restrict__ b0, const float* __restrict__ b1,
                       const float* __restrict__ b2, const float* __restrict__ b3,
                       float* __restrict__ out, int nbands) {
  const int wave = threadIdx.x >> 5;
  const int lane = threadIdx.x & 31;
  const int band = blockIdx.x * (blockDim.x >> 5) + wave;
  if (band >= nbands) return;   // wave-uniform: EXEC stays all-ones for WMMA

  const int m  = lane & 15;          // row (A/C) or col (B/C) within tile
  const int kh = (lane >> 4) << 1;   // K sub-offset: 0 or 2

  const float* aggRow = agg + ((long long)band * 16 + m) * D;
  const float* xRow   = xin + ((long long)band * 16 + m) * D;

  v8f acc[4] = {v8f{}, v8f{}, v8f{}, v8f{}};

  for (int kk = 0; kk < 16; ++kk) {
    const int k = (kk << 2) + kh;
    // A fragments (16x4 f32): lane l holds A[m][kh+v] in vgpr v -> contiguous float2
    v2f aAgg = *(const v2f*)(aggRow + k);
    v2f aX   = *(const v2f*)(xRow   + k);
#pragma unroll
    for (int jt = 0; jt < 4; ++jt) {
      const int n = (jt << 4) + m;
      // B = W^T, so B[k][n] = W[n][k]; lane holds {W[n][k], W[n][k+1]} -> float2
      v2f bRel  = *(const v2f*)(Wrel  + n * D + k);
      v2f bRoot = *(const v2f*)(Wroot + n * D + k);
      acc[jt] = __builtin_amdgcn_wmma_f32_16x16x4_f32(
          false, aAgg, false, bRel, (short)0, acc[jt], false, false);
      acc[jt] = __builtin_amdgcn_wmma_f32_16x16x4_f32(
          false, aX, false, bRoot, (short)0, acc[jt], false, false);
    }
  }

  // C/D layout: vgpr r -> row (r + 8*(lane>>4)), col (lane&15)
  const int mh = (lane >> 4) << 3;
#pragma unroll
  for (int jt = 0; jt < 4; ++jt) {
    const int n = (jt << 4) + m;
    float bias = b0[n] + b1[n];
    if (b2) bias += b2[n];
    if (b3) bias += b3[n];
#pragma unroll
    for (int r = 0; r < 8; ++r) {
      const long long row = (long long)band * 16 + mh + r;
      const float v = acc[jt][r] + bias;
      out[row * D + n] = 1.0f / (1.0f + __expf(-v));
    }
  }
}

extern "C" void kernel_launch(void* const* d_in, const int* in_sizes, int n_in,
                              void* d_out, int out_size, void* d_ws, size_t ws_size,
                              hipStream_t stream) {
  const float* X      = (const float*)d_in[0];
  const int*   ei     = (const int*)d_in[1];
  const float* ew     = (const float*)d_in[2];
  const float* Wrel1  = (const float*)d_in[3];
  const float* brel1  = (const float*)d_in[4];
  const float* Wroot1 = (const float*)d_in[5];
  const float* broot1 = (const float*)d_in[6];
  // layer 2 applies to H_prev == 0 -> contributes only brel2 + broot2
  const float* brel2  = (const float*)d_in[8];
  const float* broot2 = (const float*)d_in[10];
  const float* Wrel3  = (const float*)d_in[11];
  const float* brel3  = (const float*)d_in[12];
  const float* Wroot3 = (const float*)d_in[13];
  const float* broot3 = (const float*)d_in[14];

  const int N = in_sizes[0] / D;
  const int E = in_sizes[1] / 2;
  const int* src = ei;
  const int* dst = ei + E;

  float* agg1 = (float*)d_ws;
  float* agg3 = agg1 + (size_t)N * D;
  float* H    = agg3 + (size_t)N * D;

  // zero both accumulators up front (2*N*64 floats, divisible by 4)
  zero_kernel<<<4096, 256, 0, stream>>>((v4f*)agg1, (long long)2 * N * (D / 4));

  const long long ework = (long long)E * (D / 4);
  int sblocks = (int)((ework + 255) / 256);
  if (sblocks > 131072) sblocks = 131072;

  const int nbands = (N + 15) / 16;            // N=50000 -> 3125 exact
  const int gblocks = (nbands + 7) / 8;        // 8 waves (bands) per block

  // layer 1: agg1 = scatter_add(X[src]*ew)
  scatter_kernel<<<sblocks, 256, 0, stream>>>(X, src, dst, ew, agg1, E);
  // H = sigmoid(agg1 @ Wrel1^T + X @ Wroot1^T + brel1 + broot1 + brel2 + broot2)
  gconv_wmma_kernel<<<gblocks, 256, 0, stream>>>(agg1, X, Wrel1, Wroot1,
                                                 brel1, broot1, brel2, broot2,
                                                 H, nbands);
  // layer 3: agg3 = scatter_add(H[src]*ew)
  scatter_kernel<<<sblocks, 256, 0, stream>>>(H, src, dst, ew, agg3, E);
  // out = sigmoid(agg3 @ Wrel3^T + H @ Wroot3^T + brel3 + broot3)
  gconv_wmma_kernel<<<gblocks, 256, 0, stream>>>(agg3, H, Wrel3, Wroot3,
                                                 brel3, broot3, nullptr, nullptr,
                                                 (float*)d_out, nbands);
}